// _NonLocalBlockND_44719199486499
// MI455X (gfx1250) — compile-verified
//
#include <hip/hip_runtime.h>
#include <hip/hip_bf16.h>

typedef __attribute__((ext_vector_type(2))) float v2f;
typedef __attribute__((ext_vector_type(4))) float v4f;
typedef __attribute__((ext_vector_type(8))) float v8f;

// address-space(1) pointers: force global_load/global_store (not flat)
typedef const float __attribute__((address_space(1)))* gcf;
typedef float       __attribute__((address_space(1)))* gf;
typedef const v2f   __attribute__((address_space(1)))* gcv2;
typedef v4f         __attribute__((address_space(1)))* gv4;

__device__ __forceinline__ gcf gro(const float* p) { return (gcf)(unsigned long long)p; }
__device__ __forceinline__ gf  grw(float* p)       { return (gf)(unsigned long long)p; }

#define CC    64
#define ICC   32
#define NPIX  8192   // T*H*W
#define MPOOL 2048   // T*(H/2)*(W/2)

__device__ __forceinline__ v8f wmma_f32(v2f a, v2f b, v8f c) {
    // D = A(16x4 f32) * B(4x16 f32) + C(16x16 f32)
    return __builtin_amdgcn_wmma_f32_16x16x4_f32(
        false, a, false, b, (short)0, c, false, false);
}

// -------------------------------------------------------------------------
// Kernel 1: phi/g projections (1x1 conv as WMMA GEMM) + fused 2x2 maxpool
// grid: 64 blocks = j(4)*b(2)*t(8), 256 threads (8 waves)
// -------------------------------------------------------------------------
__global__ __launch_bounds__(256) void proj_pool_kernel(
    const float* __restrict__ x1, const float* __restrict__ x2,
    const float* __restrict__ x3, const float* __restrict__ x4,
    const float* __restrict__ phi_w, const float* __restrict__ phi_b,
    const float* __restrict__ g_w,   const float* __restrict__ g_b,
    float* __restrict__ phiOut, float* __restrict__ gOut)
{
    __shared__ float lds[ICC * 256];   // [oc][pix-in-chunk]  (32 KB)
    const int blk = blockIdx.x;
    const int t = blk & 7, b = (blk >> 3) & 1, j = blk >> 4;
    const float* xs[4] = {x1, x2, x3, x4};
    gcf x = gro(xs[j] + (size_t)b * CC * NPIX + t * 1024); // + ch*NPIX + pix

    const int tid = threadIdx.x;
    const int wave = tid >> 5, lane = tid & 31;
    const int laneHalf = lane >> 4;     // 0: K=0,1 ; 1: K=2,3
    const int lmod = lane & 15;

    for (int chunk = 0; chunk < 4; ++chunk) {       // 4 chunks of 256 pixels (8 rows)
        const int pixBaseChunk = chunk * 256;
        for (int proj = 0; proj < 2; ++proj) {      // 0 = phi, 1 = g
            gcf pw = gro((proj == 0 ? phi_w : g_w) + (size_t)j * ICC * CC);
            gcf pb = gro((proj == 0 ? phi_b : g_b) + j * ICC);
            gf  outBuf = grw(proj == 0 ? phiOut : gOut);
            __syncthreads();   // lds safe to overwrite

            // B fragments (weights) for both oc-tiles, register-resident
            v2f Bf[2][16];
            float biasv[2];
            #pragma unroll
            for (int nt = 0; nt < 2; ++nt) {
                const int oc0 = nt * 16;
                #pragma unroll
                for (int kk = 0; kk < 16; ++kk)
                    Bf[nt][kk] = *(gcv2)(pw + (oc0 + lmod) * CC + kk * 4 + laneHalf * 2);
                biasv[nt] = pb[oc0 + lmod];
            }

            for (int mi = 0; mi < 2; ++mi) {        // 16 M-tiles / 8 waves
                const int mt = wave * 2 + mi;
                const int pix0 = pixBaseChunk + mt * 16;
                v8f acc0, acc1;
                #pragma unroll
                for (int r = 0; r < 8; ++r) { acc0[r] = biasv[0]; acc1[r] = biasv[1]; }
                #pragma unroll
                for (int kk = 0; kk < 16; ++kk) {
                    const int ch = kk * 4 + laneHalf * 2;
                    v2f Af;
                    Af.x = x[(size_t)ch * NPIX + pix0 + lmod];
                    Af.y = x[(size_t)(ch + 1) * NPIX + pix0 + lmod];
                    acc0 = wmma_f32(Af, Bf[0][kk], acc0);
                    acc1 = wmma_f32(Af, Bf[1][kk], acc1);
                }
                // D: lane<16 -> rows r, lane>=16 -> rows r+8; col = oc-tile + lmod
                const int pixc = mt * 16 + laneHalf * 8;   // within chunk
                v4f lo0 = {acc0[0], acc0[1], acc0[2], acc0[3]};
                v4f hi0 = {acc0[4], acc0[5], acc0[6], acc0[7]};
                v4f lo1 = {acc1[0], acc1[1], acc1[2], acc1[3]};
                v4f hi1 = {acc1[4], acc1[5], acc1[6], acc1[7]};
                *(v4f*)(&lds[(0  + lmod) * 256 + pixc])     = lo0;
                *(v4f*)(&lds[(0  + lmod) * 256 + pixc + 4]) = hi0;
                *(v4f*)(&lds[(16 + lmod) * 256 + pixc])     = lo1;
                *(v4f*)(&lds[(16 + lmod) * 256 + pixc + 4]) = hi1;
            }
            __syncthreads();
            // 2x2 max-pool this chunk (8 rows -> 4 pooled rows): 32oc*4*16 outs
            for (int idx = tid; idx < 2048; idx += 256) {
                const int oc = idx >> 6;
                const int pr = (idx >> 4) & 3;
                const int pc = idx & 15;
                const float* base = &lds[oc * 256 + pr * 64 + pc * 2];
                const float mx = fmaxf(fmaxf(base[0], base[1]),
                                       fmaxf(base[32], base[33]));
                const int mIdx = t * 256 + (chunk * 4 + pr) * 16 + pc;
                outBuf[((size_t)(j * 2 + b) * ICC + oc) * MPOOL + mIdx] = mx;
            }
        }
    }
}

// -------------------------------------------------------------------------
// Kernel 2: S[j][b] = phi * g^T / M   (32x32, K=2048) via WMMA
// grid: 8 blocks (j*2+b), 128 threads (4 waves, one 16x16 tile each)
// -------------------------------------------------------------------------
__global__ __launch_bounds__(128) void s_kernel(
    const float* __restrict__ phi, const float* __restrict__ g,
    float* __restrict__ S)
{
    const int jb = blockIdx.x;
    gcf P = gro(phi + (size_t)jb * ICC * MPOOL);
    gcf G = gro(g   + (size_t)jb * ICC * MPOOL);
    gf  So = grw(S);
    const int tid = threadIdx.x;
    const int wave = tid >> 5, lane = tid & 31;
    const int laneHalf = lane >> 4, lmod = lane & 15;
    const int a0 = (wave >> 1) * 16, c0 = (wave & 1) * 16;

    v8f acc = {};
    for (int m0 = 0; m0 < MPOOL; m0 += 4) {
        const int mk = m0 + laneHalf * 2;
        v2f Af = *(gcv2)(P + (a0 + lmod) * MPOOL + mk);
        v2f Bf = *(gcv2)(G + (c0 + lmod) * MPOOL + mk);
        acc = wmma_f32(Af, Bf, acc);
    }
    const float scale = 1.0f / (float)MPOOL;
    #pragma unroll
    for (int r = 0; r < 8; ++r) {
        const int a = a0 + r + laneHalf * 8;
        const int c = c0 + lmod;
        So[(size_t)jb * 1024 + a * 32 + c] = acc[r] * scale;
    }
}

// -------------------------------------------------------------------------
// Kernel 3: fold theta_w, S, w_w, bias, BN into Eid = I + sum_slots E
// Eid stored TRANSPOSED as [o][c] so the GEMM B-fragments are b64 loads.
// grid: 8 blocks (i*2+b), 256 threads, scalar math (tiny).
// -------------------------------------------------------------------------
__global__ __launch_bounds__(256) void e_kernel(
    const float* __restrict__ S,
    const float* __restrict__ theta_w, const float* __restrict__ theta_b,
    const float* __restrict__ w_w, const float* __restrict__ w_b,
    const float* __restrict__ bn_gamma, const float* __restrict__ bn_beta,
    const float* __restrict__ bn_mean,  const float* __restrict__ bn_var,
    float* __restrict__ Eid, float* __restrict__ cvv)
{
    __shared__ float Sl[32 * 32];
    __shared__ float U[64 * 32];
    __shared__ float tbS[32];
    __shared__ float Et[64 * 64];
    __shared__ float cv[64];
    const int ib = blockIdx.x;
    const int b = ib & 1, i = ib >> 1;
    const int tid = threadIdx.x;
    gcf Sg  = gro(S);
    gcf twg = gro(theta_w);
    gcf tbg = gro(theta_b);
    gcf wwg = gro(w_w);
    gcf wbg = gro(w_b);
    gcf gg  = gro(bn_gamma);
    gcf bg  = gro(bn_beta);
    gcf mg  = gro(bn_mean);
    gcf vg  = gro(bn_var);

    for (int idx = tid; idx < 64 * 64; idx += 256)
        Et[idx] = ((idx >> 6) == (idx & 63)) ? 1.0f : 0.0f;
    if (tid < 64) cv[tid] = 0.0f;

    // PAIRS: 0->(1,2) 1->(0,2) 2->(1,3) 3->(2,1)
    const int t0 = (i == 0) ? 1 : (i == 1) ? 0 : (i == 2) ? 1 : 2;
    const int t1 = (i == 0) ? 2 : (i == 1) ? 2 : (i == 2) ? 3 : 1;

    for (int slot = 0; slot < 2; ++slot) {
        const int j = (slot == 0) ? t0 : t1;
        __syncthreads();
        for (int idx = tid; idx < 1024; idx += 256)
            Sl[idx] = Sg[(size_t)(j * 2 + b) * 1024 + idx];
        __syncthreads();
        // U[c][d] = sum_a theta_w[i][a][c] * S[a][d]
        for (int idx = tid; idx < 2048; idx += 256) {
            const int c = idx >> 5, d = idx & 31;
            float s = 0.f;
            for (int a = 0; a < 32; ++a)
                s += twg[((size_t)i * 32 + a) * 64 + c] * Sl[a * 32 + d];
            U[idx] = s;
        }
        if (tid < 32) {
            float s = 0.f;
            for (int a = 0; a < 32; ++a)
                s += tbg[i * 32 + a] * Sl[a * 32 + tid];
            tbS[tid] = s;
        }
        __syncthreads();
        gcf ww = wwg + (size_t)(i * 2 + slot) * 64 * 32;
        const int po = (i * 2 + slot) * 64;
        for (int idx = tid; idx < 4096; idx += 256) {
            const int c = idx >> 6, o = idx & 63;
            float e = 0.f;
            for (int d = 0; d < 32; ++d) e += U[c * 32 + d] * ww[o * 32 + d];
            const float inv = gg[po + o] / sqrtf(vg[po + o] + 1e-5f);
            Et[idx] += e * inv;
        }
        if (tid < 64) {
            const int o = tid;
            float s = 0.f;
            for (int d = 0; d < 32; ++d) s += tbS[d] * ww[o * 32 + d];
            const float inv = gg[po + o] / sqrtf(vg[po + o] + 1e-5f);
            cv[o] += (s + wbg[po + o]) * inv + bg[po + o] - mg[po + o] * inv;
        }
    }
    __syncthreads();
    gf Eo = grw(Eid);
    for (int idx = tid; idx < 4096; idx += 256) {
        const int c = idx >> 6, o = idx & 63;
        Eo[(size_t)ib * 4096 + o * 64 + c] = Et[idx];   // transposed: [o][c]
    }
    if (tid < 64) grw(cvv)[ib * 64 + tid] = cv[tid];
}

// -------------------------------------------------------------------------
// Kernel 4: Z = X * Eid + cvec  per (i,b): [8192 x 64]x[64 x 64] WMMA GEMM
// grid: 128 blocks = (i*2+b)*16 + chunk(512 pixels), 256 threads (8 waves)
// Each wave: 2 oc-tiles share one A fragment -> 2 WMMA per A load.
// -------------------------------------------------------------------------
__global__ __launch_bounds__(256) void out_kernel(
    const float* __restrict__ x1, const float* __restrict__ x2,
    const float* __restrict__ x3, const float* __restrict__ x4,
    const float* __restrict__ Eid, const float* __restrict__ cvv,
    float* __restrict__ out)
{
    const int blk = blockIdx.x;
    const int chunk = blk & 15, ib = blk >> 4;
    const int b = ib & 1, i = ib >> 1;
    const float* xs[4] = {x1, x2, x3, x4};
    const float* xgen = xs[i] + (size_t)b * CC * NPIX;
    gcf x  = gro(xgen);
    gcf E  = gro(Eid + (size_t)ib * 4096);      // [o][c] layout
    gcf cv = gro(cvv + ib * 64);
    gf  o  = grw(out + (size_t)ib * CC * NPIX);

    const int tid = threadIdx.x;
    const int wave = tid >> 5, lane = tid & 31;
    const int laneHalf = lane >> 4, lmod = lane & 15;
    const int np = wave & 1;       // oc-tile pair: {2np, 2np+1}
    const int mgrp = wave >> 1;    // quarter of the 32 pixel-tiles
    const int oc0 = np * 32;

    __builtin_prefetch(xgen + (size_t)chunk * 512, 0, 0);

    // B fragments (Eid columns, [o][c] layout -> contiguous pairs) in registers
    v2f Bf[2][16];
    float cvl[2];
    #pragma unroll
    for (int nt = 0; nt < 2; ++nt) {
        const int oc = oc0 + nt * 16 + lmod;
        #pragma unroll
        for (int kk = 0; kk < 16; ++kk)
            Bf[nt][kk] = *(gcv2)(E + oc * 64 + kk * 4 + laneHalf * 2);
        cvl[nt] = cv[oc];
    }

    for (int mi = 0; mi < 8; ++mi) {
        const int mt = mgrp * 8 + mi;
        const int pix0 = chunk * 512 + mt * 16;
        v8f acc0, acc1;
        #pragma unroll
        for (int r = 0; r < 8; ++r) { acc0[r] = cvl[0]; acc1[r] = cvl[1]; }
        #pragma unroll
        for (int kk = 0; kk < 16; ++kk) {
            const int ch = kk * 4 + laneHalf * 2;
            v2f Af;
            Af.x = x[(size_t)ch * NPIX + pix0 + lmod];
            Af.y = x[(size_t)(ch + 1) * NPIX + pix0 + lmod];
            acc0 = wmma_f32(Af, Bf[0][kk], acc0);
            acc1 = wmma_f32(Af, Bf[1][kk], acc1);
        }
        const int pix = pix0 + laneHalf * 8;
        gf dst0 = o + (size_t)(oc0 + lmod) * NPIX + pix;
        gf dst1 = o + (size_t)(oc0 + 16 + lmod) * NPIX + pix;
        v4f lo0 = {acc0[0], acc0[1], acc0[2], acc0[3]};
        v4f hi0 = {acc0[4], acc0[5], acc0[6], acc0[7]};
        v4f lo1 = {acc1[0], acc1[1], acc1[2], acc1[3]};
        v4f hi1 = {acc1[4], acc1[5], acc1[6], acc1[7]};
        *(gv4)(dst0)     = lo0;
        *(gv4)(dst0 + 4) = hi0;
        *(gv4)(dst1)     = lo1;
        *(gv4)(dst1 + 4) = hi1;
    }
}

// -------------------------------------------------------------------------
extern "C" void kernel_launch(void* const* d_in, const int* in_sizes, int n_in,
                              void* d_out, int out_size, void* d_ws, size_t ws_size,
                              hipStream_t stream) {
    const float* x1 = (const float*)d_in[0];
    const float* x2 = (const float*)d_in[1];
    const float* x3 = (const float*)d_in[2];
    const float* x4 = (const float*)d_in[3];
    const float* theta_w = (const float*)d_in[4];
    const float* theta_b = (const float*)d_in[5];
    const float* phi_w   = (const float*)d_in[6];
    const float* phi_b   = (const float*)d_in[7];
    const float* g_w     = (const float*)d_in[8];
    const float* g_b     = (const float*)d_in[9];
    const float* w_w     = (const float*)d_in[10];
    const float* w_b     = (const float*)d_in[11];
    const float* bn_gamma= (const float*)d_in[12];
    const float* bn_beta = (const float*)d_in[13];
    const float* bn_mean = (const float*)d_in[14];
    const float* bn_var  = (const float*)d_in[15];

    float* ws = (float*)d_ws;
    float* phiBuf = ws;                      // 4*2*32*2048 = 524288 floats
    float* gBuf   = phiBuf + 524288;         // 524288
    float* Sbuf   = gBuf + 524288;           // 8*1024 = 8192
    float* Ebuf   = Sbuf + 8192;             // 8*4096 = 32768
    float* cvBuf  = Ebuf + 32768;            // 8*64 = 512

    proj_pool_kernel<<<64, 256, 0, stream>>>(x1, x2, x3, x4,
                                             phi_w, phi_b, g_w, g_b,
                                             phiBuf, gBuf);
    s_kernel<<<8, 128, 0, stream>>>(phiBuf, gBuf, Sbuf);
    e_kernel<<<8, 256, 0, stream>>>(Sbuf, theta_w, theta_b, w_w, w_b,
                                    bn_gamma, bn_beta, bn_mean, bn_var,
                                    Ebuf, cvBuf);
    out_kernel<<<128, 256, 0, stream>>>(x1, x2, x3, x4, Ebuf, cvBuf,
                                        (float*)d_out);
}